// SGCN_23072564314739
// MI455X (gfx1250) — compile-verified
//
#include <hip/hip_runtime.h>
#include <hip/hip_bf16.h>

typedef __attribute__((ext_vector_type(2))) float v2f;
typedef __attribute__((ext_vector_type(8))) float v8f;

#define HDIM 64

// ---------------------------------------------------------------- utilities
__global__ void zero_f32(float* __restrict__ p, long long n) {
    long long i = (long long)blockIdx.x * blockDim.x + threadIdx.x;
    long long stride = (long long)gridDim.x * blockDim.x;
    for (; i < n; i += stride) p[i] = 0.0f;
}

// deg[r] += 1 for every edge (col, row)
__global__ void deg_accum(const int* __restrict__ edges, float* __restrict__ deg, int E) {
    int e = blockIdx.x * blockDim.x + threadIdx.x;
    if (e >= E) return;
    int r = edges[2 * e + 1];
    atomicAdd(deg + r, 1.0f);
}

__global__ void inv_deg(float* __restrict__ deg, int n) {
    int i = blockIdx.x * blockDim.x + threadIdx.x;
    if (i >= n) return;
    float d = deg[i];
    deg[i] = (d > 0.0f) ? (1.0f / d) : 0.0f;
}

// ---------------------------------------------------------------- WMMA GEMM
// Y[M x 64] = A[M x K] * W[K x 64], K % 4 == 0.
// Block = 128 threads (4 waves); block computes a 16-row x 64-col tile, wave w
// owns columns [16w, 16w+16). The 16 x K A-slab is staged into LDS with the
// gfx1250 async-copy engine (GLOBAL_LOAD_ASYNC_TO_LDS_B128, ASYNCcnt-tracked,
// no VGPR round-trip), then consumed as V_WMMA_F32_16X16X4_F32 fragments.
// ldk (LDS row stride, floats) is chosen host-side: multiple of 4 (16B async
// alignment) and ldk % 8 == 4 so the 16 fragment rows hit 16 distinct banks.
__global__ void __launch_bounds__(128)
gemm_wmma_f32(const float* __restrict__ A, const float* __restrict__ W,
              float* __restrict__ Y, int M, int K, int ldk) {
    extern __shared__ float As[];                 // 16 * ldk floats
    const int tid  = threadIdx.x;
    const int wave = tid >> 5;
    const int lane = tid & 31;
    const int m0   = blockIdx.x * 16;

    // Keep the (tiny, L2-resident) weight matrix warm: one cacheline per lane.
    for (int off = tid * 32; off < K * HDIM; off += 128 * 32)
        __builtin_prefetch(W + off, 0, 3);        // -> global_prefetch_b8

    // Async-stage the 16 x K A-slab into LDS: 16 bytes (4 floats) per lane-op.
    const int chunksPerRow = K >> 2;              // float4 chunks per row
    for (int idx = tid; idx < 16 * chunksPerRow; idx += 128) {
        int row  = idx / chunksPerRow;
        int col4 = (idx - row * chunksPerRow) << 2;
        if (m0 + row < M) {
            unsigned lds = (unsigned)(size_t)&As[row * ldk + col4]; // flat low32 = LDS addr
            unsigned long long ga =
                (unsigned long long)(const void*)(A + (size_t)(m0 + row) * K + col4);
            asm volatile("global_load_async_to_lds_b128 %0, %1, off"
                         :: "v"(lds), "v"(ga) : "memory");
        }
    }
    asm volatile("s_wait_asynccnt 0x0" ::: "memory");
    __syncthreads();

    const int n0   = wave * 16;
    const int half = lane >> 4;                   // 0: lanes 0-15, 1: lanes 16-31
    const int l15  = lane & 31 & 15;
    const int koff = half * 2;                    // A/B K-offset for upper lane half

    v8f acc = {};
    const float* ar = As + l15 * ldk + koff;      // A row for this lane
    const float* wr = W + n0 + l15;               // B column for this lane

    for (int k = 0; k < K; k += 4) {
        v2f a, b;
        // A 16x4 f32 layout: v0 = K+0 (lo half) / K+2 (hi half), v1 = K+1 / K+3
        a.x = ar[k];
        a.y = ar[k + 1];
        // B 4x16 f32 layout (symmetric): lanes index N, VGPRs index K
        b.x = wr[(size_t)(k + koff) * HDIM];
        b.y = wr[(size_t)(k + koff + 1) * HDIM];
        acc = __builtin_amdgcn_wmma_f32_16x16x4_f32(
            /*neg_a=*/false, a, /*neg_b=*/false, b,
            /*c_mod=*/(short)0, acc, /*reuse_a=*/false, /*reuse_b=*/false);
    }

    // C/D layout: VGPR j, lanes 0-15 -> (M=j, N=lane); lanes 16-31 -> (M=j+8, N=lane-16)
    #pragma unroll
    for (int j = 0; j < 8; ++j) {
        int row = m0 + j + half * 8;
        if (row < M) Y[(size_t)row * HDIM + n0 + l15] = acc[j];
    }
}

// ---------------------------------------------------------------- SpMM scatter
// Z[r, :] += Y[c, :] * invd[r] for each edge. 16 lanes per edge, 4 cols per lane.
// Gather table Y and target Z each fit in the 192MB L2 -> L2-resident traffic.
__global__ void __launch_bounds__(256)
spmm_scatter(const int* __restrict__ edges, const float* __restrict__ Y,
             const float* __restrict__ invd, float* __restrict__ Z, int E) {
    long long t = (long long)blockIdx.x * blockDim.x + threadIdx.x;
    int e = (int)(t >> 4);
    if (e >= E) return;
    int q = ((int)t & 15) * 4;
    int c = edges[2 * e];
    int r = edges[2 * e + 1];
    float s = invd[r];
    const float4 v = *reinterpret_cast<const float4*>(Y + (size_t)c * HDIM + q);
    float* zp = Z + (size_t)r * HDIM + q;
    atomicAdd(zp + 0, v.x * s);
    atomicAdd(zp + 1, v.y * s);
    atomicAdd(zp + 2, v.z * s);
    atomicAdd(zp + 3, v.w * s);
}

// ---------------------------------------------------------------- bias + leakyReLU + L2 norm
// One wave (32 lanes) per node, 2 cols per lane; shfl_xor reduction (wave32).
__global__ void __launch_bounds__(256)
post_norm(const float* __restrict__ Z, const float* __restrict__ bias,
          float* __restrict__ H, int M) {
    int wave = threadIdx.x >> 5;
    int lane = threadIdx.x & 31;
    int node = blockIdx.x * 8 + wave;
    if (node >= M) return;
    int col = lane * 2;
    float2 v = *reinterpret_cast<const float2*>(Z + (size_t)node * HDIM + col);
    v.x += bias[col];
    v.y += bias[col + 1];
    v.x = (v.x > 0.0f) ? v.x : 0.2f * v.x;
    v.y = (v.y > 0.0f) ? v.y : 0.2f * v.y;
    float ss = v.x * v.x + v.y * v.y;
    #pragma unroll
    for (int o = 16; o > 0; o >>= 1) ss += __shfl_xor(ss, o, 32);
    float scale = 1.0f / fmaxf(sqrtf(ss), 1e-12f);
    float2 out;
    out.x = v.x * scale;
    out.y = v.y * scale;
    *reinterpret_cast<float2*>(H + (size_t)node * HDIM + col) = out;
}

// ---------------------------------------------------------------- row gather
__global__ void gather_rows(const float* __restrict__ H, const int* __restrict__ idx,
                            float* __restrict__ out, int nlab) {
    int t = blockIdx.x * blockDim.x + threadIdx.x;
    if (t >= nlab * HDIM) return;
    int i = t >> 6;
    int j = t & 63;
    out[t] = H[(size_t)idx[i] * HDIM + j];
}

// ---------------------------------------------------------------- launcher
extern "C" void kernel_launch(void* const* d_in, const int* in_sizes, int n_in,
                              void* d_out, int out_size, void* d_ws, size_t ws_size,
                              hipStream_t stream) {
    const int*   edges = (const int*)d_in[0];    // (E,2) int32: [:,0]=col, [:,1]=row
    const float* feat  = (const float*)d_in[1];  // (N,300)
    const int*   lidx  = (const int*)d_in[2];    // (1000,)
    const float* w1    = (const float*)d_in[3];  // (300,64)
    const float* b1    = (const float*)d_in[4];  // (64,)
    const float* w2    = (const float*)d_in[5];  // (64,64)
    const float* b2    = (const float*)d_in[6];  // (64,)

    const int E    = in_sizes[0] / 2;
    const int K1   = 300;
    const int M    = in_sizes[1] / K1;           // 100000 nodes
    const int nlab = in_sizes[2];

    // LDS row strides: multiple of 4 floats (16B async alignment) and
    // stride % 8 == 4 (=> stride mod 64 = 4*odd => 16 conflict-free banks).
    const int ldk1 = (K1 % 8 == 4) ? K1 : K1 + 4;      // 300 -> 300
    const int ldk2 = (HDIM % 8 == 4) ? HDIM : HDIM + 4; // 64 -> 68

    // Workspace carve-up (~77 MB total)
    float* deg  = (float*)d_ws;                  // M floats (deg -> inv_deg in place)
    float* bufA = deg  + M;                      // M x 64: GEMM output
    float* bufB = bufA + (size_t)M * HDIM;       // M x 64: SpMM accumulator
    float* bufC = bufB + (size_t)M * HDIM;       // M x 64: normalized hidden

    const int T = 256;
    const long long mh = (long long)M * HDIM;

    // degrees
    zero_f32<<<2048, T, 0, stream>>>(deg, M);
    zero_f32<<<2048, T, 0, stream>>>(bufB, mh);
    deg_accum<<<(E + T - 1) / T, T, 0, stream>>>(edges, deg, E);
    inv_deg<<<(M + T - 1) / T, T, 0, stream>>>(deg, M);

    // ---- layer 1 ----
    gemm_wmma_f32<<<(M + 15) / 16, 128, (size_t)16 * ldk1 * sizeof(float), stream>>>(
        feat, w1, bufA, M, K1, ldk1);
    spmm_scatter<<<(int)(((long long)E * 16 + T - 1) / T), T, 0, stream>>>(
        edges, bufA, deg, bufB, E);
    post_norm<<<(M + 7) / 8, T, 0, stream>>>(bufB, b1, bufC, M);

    // ---- layer 2 ----
    zero_f32<<<2048, T, 0, stream>>>(bufB, mh);
    gemm_wmma_f32<<<(M + 15) / 16, 128, (size_t)16 * ldk2 * sizeof(float), stream>>>(
        bufC, w2, bufA, M, HDIM, ldk2);
    spmm_scatter<<<(int)(((long long)E * 16 + T - 1) / T), T, 0, stream>>>(
        edges, bufA, deg, bufB, E);
    post_norm<<<(M + 7) / 8, T, 0, stream>>>(bufB, b2, bufC, M);

    // ---- output gather ----
    gather_rows<<<(nlab * HDIM + T - 1) / T, T, 0, stream>>>(bufC, lidx, (float*)d_out, nlab);
}